// RecurrentInferenceMachine_25048249270795
// MI455X (gfx1250) — compile-verified
//
#include <hip/hip_runtime.h>

// Problem constants (B=1)
static constexpr int TT    = 12;
static constexpr int DD    = 96;
static constexpr int WWp   = 96;
static constexpr int PLANE = DD * WWp;      // 9216
static constexpr int NP    = TT * PLANE;    // 110592
static constexpr int FC    = 64;

typedef __attribute__((ext_vector_type(2))) float v2f;
typedef __attribute__((ext_vector_type(8))) float v8f;

// fp32 WMMA: D(16x16 f32) = A(16x4 f32) * B(4x16 f32) + C
__device__ __forceinline__ v8f wmma4(v2f a, v2f b, v8f c) {
  return __builtin_amdgcn_wmma_f32_16x16x4_f32(false, a, false, b, (short)0, c,
                                               false, false);
}

__device__ __forceinline__ int clampi(int v, int lo, int hi) {
  return v < lo ? lo : (v > hi ? hi : v);
}

__device__ __forceinline__ v2f ld2(const float* p) {
  return *reinterpret_cast<const v2f*>(p);   // 8B-aligned by construction
}

// ---------------------------------------------------------------------------
// Weight repack: (O=64, I=64, ntap) -> (ntap, O, I) so WMMA A-fragments
// (rows o, cols i,i+1) become single aligned b64 loads.
// ---------------------------------------------------------------------------
__global__ void k_repack(const float* __restrict__ w, float* __restrict__ wt,
                         int ntap) {
  int idx = blockIdx.x * blockDim.x + threadIdx.x;
  int total = FC * FC * ntap;
  if (idx >= total) return;
  int tap = idx / (FC * FC);
  int r   = idx - tap * FC * FC;
  int o   = r / FC;
  int i   = r - o * FC;
  wt[idx] = w[(o * FC + i) * ntap + tap];
}

// ---------------------------------------------------------------------------
// Input conv: 4->64 channels, 3x3 over (D,W) with edge padding, bias+relu.
// K=36 is too shallow for WMMA tiles; direct VALU with full input reuse.
// ---------------------------------------------------------------------------
__global__ void k_conv_in(const float* __restrict__ x,
                          const float* __restrict__ wgt,   // (64,4,1,3,3)
                          const float* __restrict__ bias,  // (64)
                          float* __restrict__ out) {       // (64,NP)
  int n = blockIdx.x * blockDim.x + threadIdx.x;
  if (n >= NP) return;
  int t = n / PLANE;
  int r = n - t * PLANE;
  int d = r / WWp;
  int w = r - d * WWp;

  float xv[36];
#pragma unroll
  for (int i = 0; i < 4; ++i)
#pragma unroll
    for (int dy = -1; dy <= 1; ++dy)
#pragma unroll
      for (int dx = -1; dx <= 1; ++dx)
        xv[i * 9 + (dy + 1) * 3 + (dx + 1)] =
            x[i * NP + t * PLANE + clampi(d + dy, 0, DD - 1) * WWp +
              clampi(w + dx, 0, WWp - 1)];

  for (int o = 0; o < FC; ++o) {
    float acc = bias[o];
#pragma unroll
    for (int j = 0; j < 36; ++j) acc = fmaf(wgt[o * 36 + j], xv[j], acc);
    out[o * NP + n] = acc > 0.f ? acc : 0.f;
  }
}

// ---------------------------------------------------------------------------
// Spatial conv 64->64, 3x3 over (D,W), edge pad, bias+relu. Implicit GEMM:
// 9 accumulated GEMMs (one per tap) of K=64 via WMMA f32 16x16x4.
// One wave = one 16-pixel N tile (W=96 = 6*16, tiles never straddle rows)
// x all 4 M tiles; B fragments shared across M tiles; repacked (tap,O,I)
// weights make each A fragment a single b64 load.
// ---------------------------------------------------------------------------
__global__ void k_conv_spatial64(const float* __restrict__ in,   // (64,NP)
                                 const float* __restrict__ wt,   // (9,64,64)
                                 const float* __restrict__ bias, // (64)
                                 float* __restrict__ out) {      // (64,NP)
  const int lane = threadIdx.x & 31;
  const int half = lane >> 4;   // K-pair select
  const int nl   = lane & 15;   // M row (A) / N col (B,C)

  int tile = blockIdx.x * 4 + (threadIdx.x >> 5);   // 6912 tiles total
  int t   = tile / 576;                              // 576 tiles per t-plane
  int rem = tile - t * 576;
  int d   = rem / 6;
  int w0  = (rem - d * 6) * 16;

  v8f acc[4];
#pragma unroll
  for (int m = 0; m < 4; ++m) acc[m] = {};

  for (int dy = -1; dy <= 1; ++dy) {
    int dsrc = clampi(d + dy, 0, DD - 1);
    for (int dx = -1; dx <= 1; ++dx) {
      int wsrc = clampi(w0 + nl + dx, 0, WWp - 1);
      const float* src = in + t * PLANE + dsrc * WWp + wsrc;
      // Preload B fragments for this tap (shared across all 4 M tiles)
      v2f bf[16];
#pragma unroll
      for (int kk = 0; kk < 16; ++kk) {
        int k0 = 4 * kk + 2 * half;
        bf[kk].x = src[k0 * NP];
        bf[kk].y = src[(k0 + 1) * NP];
      }
      const float* wtap = wt + ((dy + 1) * 3 + (dx + 1)) * FC * FC;
      for (int mt = 0; mt < 4; ++mt) {
        const float* wrow = wtap + (mt * 16 + nl) * FC;
#pragma unroll
        for (int kk = 0; kk < 16; ++kk) {
          acc[mt] = wmma4(ld2(wrow + 4 * kk + 2 * half), bf[kk], acc[mt]);
        }
      }
    }
  }

  float* dst = out + t * PLANE + d * WWp + w0 + nl;
  for (int mt = 0; mt < 4; ++mt) {
#pragma unroll
    for (int r = 0; r < 8; ++r) {
      int c = mt * 16 + r + 8 * half;
      float v = acc[mt][r] + bias[c];
      dst[c * NP] = v > 0.f ? v : 0.f;
    }
  }
}

// ---------------------------------------------------------------------------
// Temporal conv 64->64, 3 taps over T with wrap padding, bias+relu.
// 3 accumulated GEMMs of K=64 via WMMA f32 16x16x4; repacked (3,O,I) weights.
// ---------------------------------------------------------------------------
__global__ void k_conv_temporal64(const float* __restrict__ in,   // (64,NP)
                                  const float* __restrict__ wt,   // (3,64,64)
                                  const float* __restrict__ bias, // (64)
                                  float* __restrict__ out) {      // (64,NP)
  const int lane = threadIdx.x & 31;
  const int half = lane >> 4;
  const int nl   = lane & 15;

  int tile = blockIdx.x * 4 + (threadIdx.x >> 5);
  int t  = tile / 576;
  int p0 = (tile - t * 576) * 16;

  v8f acc[4];
#pragma unroll
  for (int m = 0; m < 4; ++m) acc[m] = {};

  for (int dt = 0; dt < 3; ++dt) {
    int ts = (t + dt + TT - 1) % TT;   // wrap pad
    const float* src = in + ts * PLANE + p0 + nl;
    v2f bf[16];
#pragma unroll
    for (int kk = 0; kk < 16; ++kk) {
      int k0 = 4 * kk + 2 * half;
      bf[kk].x = src[k0 * NP];
      bf[kk].y = src[(k0 + 1) * NP];
    }
    const float* wtap = wt + dt * FC * FC;
    for (int mt = 0; mt < 4; ++mt) {
      const float* wrow = wtap + (mt * 16 + nl) * FC;
#pragma unroll
      for (int kk = 0; kk < 16; ++kk) {
        acc[mt] = wmma4(ld2(wrow + 4 * kk + 2 * half), bf[kk], acc[mt]);
      }
    }
  }

  float* dst = out + t * PLANE + p0 + nl;
  for (int mt = 0; mt < 4; ++mt) {
#pragma unroll
    for (int r = 0; r < 8; ++r) {
      int c = mt * 16 + r + 8 * half;
      float v = acc[mt][r] + bias[c];
      dst[c * NP] = v > 0.f ? v : 0.f;
    }
  }
}

// ---------------------------------------------------------------------------
// Fused LRU: gi = Wih(384x64)@X + bih ; gh = Whh(384x64)@H ; GRU-style gating.
// Wave handles one 16-pixel N tile x one 16-channel tile: 12 gate tiles stay
// in accumulator registers; X/H B-fragments preloaded once and reused by all
// 6 gates. Gating runs elementwise directly in the WMMA C layout. Weight rows
// are contiguous (384x64 row-major), so A fragments are single b64 loads.
// ---------------------------------------------------------------------------
__global__ void k_lru(const float* __restrict__ X,    // (64,NP) LRU input
                      const float* __restrict__ H,    // (64,NP) hidden state
                      const float* __restrict__ wih,  // (384,64)
                      const float* __restrict__ bih,  // (384)
                      const float* __restrict__ whh,  // (384,64)
                      const float* __restrict__ bh,   // (64)
                      float* __restrict__ out,        // (64,NP)
                      float* __restrict__ hnew) {     // (64,NP)
  const int lane = threadIdx.x & 31;
  const int half = lane >> 4;
  const int nl   = lane & 15;
  const int ct   = threadIdx.x >> 5;   // channel tile 0..3
  const int nb   = blockIdx.x * 16;    // pixel base

  // Preload B fragments for X and H (reused across all 6 gates x 2 GEMMs)
  v2f bx[16], bhf[16];
#pragma unroll
  for (int kk = 0; kk < 16; ++kk) {
    int k0 = 4 * kk + 2 * half;
    bx[kk].x  = X[k0 * NP + nb + nl];
    bx[kk].y  = X[(k0 + 1) * NP + nb + nl];
    bhf[kk].x = H[k0 * NP + nb + nl];
    bhf[kk].y = H[(k0 + 1) * NP + nb + nl];
  }

  // gate order (rows of the 384): 0=ii_r 1=ih_r 2=iz 3=hz 4=ii_c 5=ih_c
  v8f s[4];               // sigmoid(reset_i, reset_h, update_i, update_h)
  v8f gi4 = {}, gh4 = {}, gi5 = {}, gh5 = {};

  for (int g = 0; g < 6; ++g) {
    int mbase = g * FC + ct * 16;
    const float* wi = wih + (mbase + nl) * FC;
    const float* wh = whh + (mbase + nl) * FC;
    v8f ai = {}, ah = {};
#pragma unroll
    for (int kk = 0; kk < 16; ++kk)
      ai = wmma4(ld2(wi + 4 * kk + 2 * half), bx[kk], ai);
#pragma unroll
    for (int kk = 0; kk < 16; ++kk)
      ah = wmma4(ld2(wh + 4 * kk + 2 * half), bhf[kk], ah);
#pragma unroll
    for (int r = 0; r < 8; ++r) ai[r] += bih[mbase + r + 8 * half];
    if (g < 4) {
#pragma unroll
      for (int r = 0; r < 8; ++r) {
        float u = ai[r] + ah[r];
        s[g][r] = 1.f / (1.f + __expf(-u));
      }
    } else if (g == 4) { gi4 = ai; gh4 = ah; }
    else               { gi5 = ai; gh5 = ah; }
  }

#pragma unroll
  for (int r = 0; r < 8; ++r) {
    int c = ct * 16 + r + 8 * half;
    int idx = c * NP + nb + nl;
    float xv = X[idx], hv = H[idx], bhv = bh[c];
    float cand_i = tanhf(gi5[r] + s[1][r] * (gh5[r] + bhv));
    float cand_h = tanhf(s[0][r] * gi4[r] + gh4[r]);
    out[idx]  = s[2][r] * cand_h + (1.f - s[2][r]) * xv;
    hnew[idx] = s[3][r] * cand_i + (1.f - s[3][r]) * hv;
  }
}

// ---------------------------------------------------------------------------
// Output head: temporal 3-tap 64->2, wrap pad, bias, NO relu. Tiny (2 rows):
// direct VALU, 384 FMAs/pixel.
// ---------------------------------------------------------------------------
__global__ void k_conv_out(const float* __restrict__ in,    // (64,NP)
                           const float* __restrict__ wgt,   // (2,64,3,1,1)
                           const float* __restrict__ bias,  // (2)
                           float* __restrict__ eta) {       // (2,NP)
  int n = blockIdx.x * blockDim.x + threadIdx.x;
  if (n >= NP) return;
  int t = n / PLANE;
  int p = n - t * PLANE;
  for (int o = 0; o < 2; ++o) {
    float acc = bias[o];
    for (int dt = 0; dt < 3; ++dt) {
      int ts = (t + dt + TT - 1) % TT;
      const float* src = in + ts * PLANE + p;
      const float* wr  = wgt + (o * FC) * 3 + dt;
#pragma unroll 8
      for (int i = 0; i < FC; ++i) acc = fmaf(wr[i * 3], src[i * NP], acc);
    }
    eta[o * NP + n] = acc;
  }
}

// ---------------------------------------------------------------------------
extern "C" void kernel_launch(void* const* d_in, const int* in_sizes, int n_in,
                              void* d_out, int out_size, void* d_ws,
                              size_t ws_size, hipStream_t stream) {
  (void)in_sizes; (void)n_in; (void)out_size; (void)ws_size;

  const float* x       = (const float*)d_in[0];
  const float* h1      = (const float*)d_in[1];
  const float* h2      = (const float*)d_in[2];
  const float* win_w1  = (const float*)d_in[3];
  const float* win_b1  = (const float*)d_in[4];
  const float* win_w2  = (const float*)d_in[5];
  const float* win_b2  = (const float*)d_in[6];
  const float* r1_wih  = (const float*)d_in[7];
  const float* r1_bih  = (const float*)d_in[8];
  const float* r1_whh  = (const float*)d_in[9];
  const float* r1_bh   = (const float*)d_in[10];
  const float* wbt_w1  = (const float*)d_in[11];
  const float* wbt_b1  = (const float*)d_in[12];
  const float* wbt_w2  = (const float*)d_in[13];
  const float* wbt_b2  = (const float*)d_in[14];
  const float* r2_wih  = (const float*)d_in[15];
  const float* r2_bih  = (const float*)d_in[16];
  const float* r2_whh  = (const float*)d_in[17];
  const float* r2_bh   = (const float*)d_in[18];
  const float* wout_w1 = (const float*)d_in[19];
  const float* wout_b1 = (const float*)d_in[20];
  const float* wout_w2 = (const float*)d_in[21];
  const float* wout_b2 = (const float*)d_in[22];

  float* eta  = (float*)d_out;                 // 2*NP
  float* h1n  = eta + 2 * NP;                  // 64*NP
  float* h2n  = h1n + FC * NP;                 // 64*NP

  // Workspace layout: two ping-pong activation buffers + repacked weights
  float* bufA    = (float*)d_ws;               // 64*NP fp32 (28.3 MB)
  float* bufB    = bufA + (size_t)FC * NP;     // 64*NP fp32
  float* wtmp    = bufB + (size_t)FC * NP;
  float* wbtS    = wtmp;                       // (9,64,64) wbt_w1 repacked
  float* woutS   = wbtS  + 9 * FC * FC;        // (9,64,64) wout_w1 repacked
  float* winT    = woutS + 9 * FC * FC;        // (3,64,64) win_w2 repacked
  float* wbtT    = winT  + 3 * FC * FC;        // (3,64,64) wbt_w2 repacked

  const int PIX_BLOCKS  = (NP + 255) / 256;    // 432
  const int WMMA_BLOCKS = (NP / 16) / 4;       // 1728 (4 waves/block)
  const int LRU_BLOCKS  = NP / 16;             // 6912 (4 channel-tile waves)
  const int RPK9        = (9 * FC * FC + 255) / 256;
  const int RPK3        = (3 * FC * FC + 255) / 256;

  // One-time weight repacks (tap-major => b64 A-fragment loads)
  k_repack<<<RPK9, 256, 0, stream>>>(wbt_w1,  wbtS,  9);
  k_repack<<<RPK9, 256, 0, stream>>>(wout_w1, woutS, 9);
  k_repack<<<RPK3, 256, 0, stream>>>(win_w2,  winT,  3);
  k_repack<<<RPK3, 256, 0, stream>>>(wbt_w2,  wbtT,  3);

  // block 1: conv_in -> temporal -> LRU1
  k_conv_in        <<<PIX_BLOCKS, 256, 0, stream>>>(x, win_w1, win_b1, bufA);
  k_conv_temporal64<<<WMMA_BLOCKS, 128, 0, stream>>>(bufA, winT, win_b2, bufB);
  k_lru            <<<LRU_BLOCKS, 128, 0, stream>>>(bufB, h1, r1_wih, r1_bih,
                                                    r1_whh, r1_bh, bufA, h1n);
  // block 2: spatial -> temporal -> LRU2
  k_conv_spatial64 <<<WMMA_BLOCKS, 128, 0, stream>>>(bufA, wbtS, wbt_b1, bufB);
  k_conv_temporal64<<<WMMA_BLOCKS, 128, 0, stream>>>(bufB, wbtT, wbt_b2, bufA);
  k_lru            <<<LRU_BLOCKS, 128, 0, stream>>>(bufA, h2, r2_wih, r2_bih,
                                                    r2_whh, r2_bh, bufB, h2n);
  // output head: spatial -> temporal 64->2
  k_conv_spatial64 <<<WMMA_BLOCKS, 128, 0, stream>>>(bufB, woutS, wout_b1, bufA);
  k_conv_out       <<<PIX_BLOCKS, 256, 0, stream>>>(bufA, wout_w2, wout_b2, eta);
}